// SpatialGraphConvBlock_17325898072117
// MI455X (gfx1250) — compile-verified
//
#include <hip/hip_runtime.h>

typedef __attribute__((ext_vector_type(2))) float v2f;
typedef __attribute__((ext_vector_type(8))) float v8f;

#define NB     64
#define CIN    64
#define COUT   64
#define TDIM   300
#define VDIM   25
#define TTILE  2          // t's per block (300 % 2 == 0 -> no t bounds checks)

#define XS_STRIDE 60      // x tile row stride (floats): cols = t*28 + v, v padded to 28
#define XA_STRIDE 74      // xa tile row stride (floats): cols = t*32 + w, w padded to 32

// Fused spatial-graph-conv:  y = W @ (x @ A_hat) + b, both matmuls on v_wmma_f32_16x16x4_f32.
__global__ __launch_bounds__(128)
void stgcn_fused_wmma(const float* __restrict__ x,
                      const float* __restrict__ Ahat,
                      const float* __restrict__ W,
                      const float* __restrict__ bias,
                      float* __restrict__ y) {
    __shared__ float xs[64 * XS_STRIDE];   // x tile  [c][t*28+v]   (15.4 KB)
    __shared__ float xas[64 * XA_STRIDE];  // xa tile [c][t*32+w]   (18.9 KB)

    const int tid  = threadIdx.x;
    const int lane = tid & 31;
    const int wave = tid >> 5;        // 0..3 : owns one 16-row M-tile in both stages
    const int half = lane >> 4;       // 0/1  : K-pair / D-row-half selector
    const int l16  = lane & 15;

    const int tt = blockIdx.x;        // 0..149
    const int n  = blockIdx.y;        // 0..63
    const int t0 = tt * TTILE;

    // ---- preload invariant WMMA fragments into registers ----
    // Stage-2 A-operand: W rows o = wave*16 + l16, K-pairs over C. (16 k-steps)
    v2f wfrag[16];
    {
        const int o = wave * 16 + l16;
        #pragma unroll
        for (int k = 0; k < 16; ++k) {
            const int kk = 4 * k + 2 * half;
            const float* p = W + o * CIN + kk;
            wfrag[k].x = p[0];
            wfrag[k].y = p[1];
        }
    }
    // Stage-1 B-operand: A_hat zero-padded to 28x32. (2 N-tiles x 7 k-steps)
    v2f afrag[2][7];
    #pragma unroll
    for (int nt = 0; nt < 2; ++nt) {
        const int w = nt * 16 + l16;
        #pragma unroll
        for (int k = 0; k < 7; ++k) {
            const int kk = 4 * k + 2 * half;
            afrag[nt][k].x = (kk     < VDIM && w < VDIM) ? Ahat[kk       * VDIM + w] : 0.0f;
            afrag[nt][k].y = (kk + 1 < VDIM && w < VDIM) ? Ahat[(kk + 1) * VDIM + w] : 0.0f;
        }
    }
    // Bias per stage-2 D row.
    float bfrag[8];
    #pragma unroll
    for (int g = 0; g < 8; ++g) bfrag[g] = bias[wave * 16 + g + 8 * half];

    // ---- cooperative load of the x tile (zero-fill v = 25..27 K-padding) ----
    {
        const float* xn = x + (size_t)n * CIN * (TDIM * VDIM);
        for (int idx = tid; idx < 64 * (TTILE * 28); idx += 128) {
            const int c = idx / (TTILE * 28);
            const int r = idx % (TTILE * 28);
            const int t = r / 28, v = r % 28;
            float val = 0.0f;
            if (v < VDIM)
                val = xn[(size_t)c * (TDIM * VDIM) + (size_t)(t0 + t) * VDIM + v];
            xs[c * XS_STRIDE + t * 28 + v] = val;
        }
    }
    __syncthreads();

    // ---- stage 1: xa[c,t,w] = sum_v x[c,t,v] * A_hat[v,w]   (WMMA, M=c K=v N=w) ----
    {
        const int c_base = wave * 16;
        #pragma unroll
        for (int t = 0; t < TTILE; ++t) {
            #pragma unroll
            for (int nt = 0; nt < 2; ++nt) {
                v8f acc = {};
                #pragma unroll
                for (int k = 0; k < 7; ++k) {
                    const int kk = 4 * k + 2 * half;
                    const float* p = &xs[(c_base + l16) * XS_STRIDE + t * 28 + kk];
                    v2f a;
                    a.x = p[0];
                    a.y = p[1];
                    acc = __builtin_amdgcn_wmma_f32_16x16x4_f32(
                        false, a, false, afrag[nt][k], (short)0, acc, false, false);
                }
                // scatter D tile into xa LDS (pad cols w>=25 hold exact zeros)
                #pragma unroll
                for (int g = 0; g < 8; ++g) {
                    const int c   = c_base + g + 8 * half;
                    const int col = t * 32 + nt * 16 + l16;
                    xas[c * XA_STRIDE + col] = acc[g];
                }
            }
        }
    }
    __syncthreads();

    // ---- stage 2: y[o,pos] = sum_c W[o,c] * xa[c,pos] + b[o]  (WMMA, M=o K=c N=pos) ----
    {
        float* yn = y + (size_t)n * COUT * (TDIM * VDIM);
        #pragma unroll
        for (int nt = 0; nt < 4; ++nt) {
            const int p0 = nt * 16;
            v8f acc = {};
            #pragma unroll
            for (int k = 0; k < 16; ++k) {
                const int kk = 4 * k + 2 * half;
                v2f bmat;
                bmat.x = xas[kk       * XA_STRIDE + p0 + l16];
                bmat.y = xas[(kk + 1) * XA_STRIDE + p0 + l16];
                acc = __builtin_amdgcn_wmma_f32_16x16x4_f32(
                    false, wfrag[k], false, bmat, (short)0, acc, false, false);
            }
            const int pos = p0 + l16;
            const int t   = pos >> 5;   // pos / 32
            const int w   = pos & 31;   // pos % 32
            if (w < VDIM) {
                #pragma unroll
                for (int g = 0; g < 8; ++g) {
                    const int o = wave * 16 + g + 8 * half;
                    yn[(size_t)o * (TDIM * VDIM) + (size_t)(t0 + t) * VDIM + w] =
                        acc[g] + bfrag[g];
                }
            }
        }
    }
}

extern "C" void kernel_launch(void* const* d_in, const int* in_sizes, int n_in,
                              void* d_out, int out_size, void* d_ws, size_t ws_size,
                              hipStream_t stream) {
    (void)in_sizes; (void)n_in; (void)out_size; (void)d_ws; (void)ws_size;
    const float* x    = (const float*)d_in[0];   // (64, 64, 300, 25)
    const float* Ahat = (const float*)d_in[1];   // (25, 25)
    const float* W    = (const float*)d_in[2];   // (64, 64)
    const float* b    = (const float*)d_in[3];   // (64,)
    float* y          = (float*)d_out;           // (64, 64, 300, 25)

    dim3 grid(TDIM / TTILE, NB);  // (150, 64)
    dim3 block(128);              // 4 wave32s
    stgcn_fused_wmma<<<grid, block, 0, stream>>>(x, Ahat, W, b, y);
}